// LocallyConnected1D_49005576847840
// MI455X (gfx1250) — compile-verified
//
#include <hip/hip_runtime.h>
#include <stdint.h>

// LocallyConnected1D: out[b,l,f] = sum_k patches[b,l,k]*kernel[l,k,f] + bias[l,f]
// B=64, L=256, C=192, F=192, KSZ=3 -> out_len=254, K=576. All fp32.
// One workgroup per l; per-l GEMM M=64,N=192,K=576 via V_WMMA_F32_16X16X4_F32.
// K streamed through double-buffered LDS using GLOBAL_LOAD_ASYNC_TO_LDS_B128
// (ASYNCcnt-tracked DMA, no VGPR staging), published with s_wait_asynccnt+barrier.

typedef __attribute__((ext_vector_type(2))) float v2f;
typedef __attribute__((ext_vector_type(8))) float v8f;

#define BATCH   64
#define LIN     256
#define CIN     192
#define FOUT    192
#define OUT_LEN 254
#define KTOT    576            // 3*192
#define CHUNK   16             // K per LDS chunk
#define NCHUNK  (KTOT / CHUNK) // 36 (even)
#define APAD    20             // padded A row (16 K + 4): conflict-free b64 frag reads
#define BPAD    208            // padded B row (192 N + 16): K-halves hit disjoint banks

__global__ __launch_bounds__(256, 2)
void lc1d_wmma_f32(const float* __restrict__ x,
                   const float* __restrict__ kern,
                   const float* __restrict__ bias,
                   float* __restrict__ out) {
  __shared__ __align__(16) float Al[2][BATCH * APAD];   // patches: [m][k] padded
  __shared__ __align__(16) float Bl[2][CHUNK * BPAD];   // weights: [k][n] padded

  const int l    = blockIdx.x;
  const int tid  = threadIdx.x;
  const int wave = tid >> 5;
  const int lane = tid & 31;
  const int h    = lane >> 4;   // half-wave select (K-pair / M+8)
  const int ln   = lane & 15;
  const int mh   = wave & 1;    // m-tile pair: tiles {2*mh, 2*mh+1}
  const int ng   = wave >> 1;   // n-tile group: tiles {3*ng .. 3*ng+2}

  const float* slab = kern + (size_t)l * KTOT * FOUT;   // this l's 576x192 weights

  v8f acc[2][3] = {};

  // Issue ASYNCcnt-tracked global->LDS copies for one K-chunk.
  // Low 32 bits of a generic pointer to __shared__ == wave-relative LDS offset.
  auto asyncStage = [&](int it, int buf) {
    const int k0 = it * CHUNK;
    const float* bsrc = slab + (size_t)k0 * FOUT;   // 16x192 chunk, contiguous
#pragma unroll
    for (int r = 0; r < 3; ++r) {
      const int idx = tid + 256 * r;
      const int fi  = idx * 4;          // flat float index within chunk
      const int k   = fi / FOUT;
      const int n   = fi % FOUT;        // multiple of 4
      unsigned lds          = (unsigned)(uintptr_t)&Bl[buf][k * BPAD + n];
      unsigned long long ga = (unsigned long long)(uintptr_t)(bsrc + fi);
      asm volatile("global_load_async_to_lds_b128 %0, %1, off"
                   :: "v"(lds), "v"(ga) : "memory");
    }
    // A chunk: x[m][l+row][c0..c0+16), row = k0/192, c0 = k0%192 (16 | 192)
    const int row = k0 / CIN;
    const int c0  = k0 % CIN;
    const int m   = tid >> 2;
    const int q   = tid & 3;
    unsigned lds          = (unsigned)(uintptr_t)&Al[buf][m * APAD + 4 * q];
    unsigned long long ga = (unsigned long long)(uintptr_t)
        (x + ((size_t)m * LIN + (l + row)) * CIN + c0 + 4 * q);
    asm volatile("global_load_async_to_lds_b128 %0, %1, off"
                 :: "v"(lds), "v"(ga) : "memory");
  };

  auto waitAsync = [&]() {
#if defined(__has_builtin)
#if __has_builtin(__builtin_amdgcn_s_wait_asynccnt)
    __builtin_amdgcn_s_wait_asynccnt(0);
#else
    asm volatile("s_wait_asynccnt 0x0" ::: "memory");
#endif
#else
    asm volatile("s_wait_asynccnt 0x0" ::: "memory");
#endif
  };

  auto compute = [&](int buf) {
#pragma unroll
    for (int s = 0; s < 4; ++s) {            // 4 K-substeps of 4
      const int kk = 4 * s + 2 * h;          // this half-wave's K pair
      v2f a[2], b[3];
#pragma unroll
      for (int mi = 0; mi < 2; ++mi) {
        const int m = (2 * mh + mi) * 16 + ln;
        a[mi] = *(const v2f*)&Al[buf][m * APAD + kk];          // ds_load_b64
      }
#pragma unroll
      for (int ni = 0; ni < 3; ++ni) {
        const int n = (3 * ng + ni) * 16 + ln;
        v2f bv;
        bv.x = Bl[buf][kk * BPAD + n];
        bv.y = Bl[buf][(kk + 1) * BPAD + n];
        b[ni] = bv;
      }
#pragma unroll
      for (int mi = 0; mi < 2; ++mi)
#pragma unroll
        for (int ni = 0; ni < 3; ++ni)
          acc[mi][ni] = __builtin_amdgcn_wmma_f32_16x16x4_f32(
              false, a[mi], false, b[ni], (short)0, acc[mi][ni], false, false);
    }
  };

  // Prologue: DMA chunk 0 into buffer 0, publish.
  asyncStage(0, 0);
  waitAsync();
  __syncthreads();

  // Main loop, unrolled x2 so buffer indices are compile-time constants.
  // NCHUNK == 36 (even): the (it+1) chunk always exists.
  for (int it = 0; it < NCHUNK; it += 2) {
    asyncStage(it + 1, 1);      // DMA next chunk while computing this one
    compute(0);                 // 24 WMMAs on buffer 0
    waitAsync();
    __syncthreads();

    if (it + 2 < NCHUNK) asyncStage(it + 2, 0);
    compute(1);                 // 24 WMMAs on buffer 1
    waitAsync();
    __syncthreads();
  }

  // Epilogue: + bias, scatter to out[b][l][f]
  const float* bs = bias + (size_t)l * FOUT;
#pragma unroll
  for (int ni = 0; ni < 3; ++ni) {
    const int n    = (3 * ng + ni) * 16 + ln;
    const float bv = bs[n];
#pragma unroll
    for (int mi = 0; mi < 2; ++mi) {
      const int mbase = (2 * mh + mi) * 16 + 8 * h;   // C/D layout: VGPR j -> M = j + 8h
#pragma unroll
      for (int j = 0; j < 8; ++j) {
        const int bb = mbase + j;
        out[((size_t)bb * OUT_LEN + l) * FOUT + n] = acc[mi][ni][j] + bv;
      }
    }
  }
}

extern "C" void kernel_launch(void* const* d_in, const int* in_sizes, int n_in,
                              void* d_out, int out_size, void* d_ws, size_t ws_size,
                              hipStream_t stream) {
  const float* x    = (const float*)d_in[0];
  const float* kern = (const float*)d_in[1];
  const float* bias = (const float*)d_in[2];
  float* out        = (float*)d_out;

  dim3 grid(OUT_LEN);   // one workgroup per output position l
  dim3 block(256);      // 8 wave32 waves
  hipLaunchKernelGGL(lc1d_wmma_f32, grid, block, 0, stream, x, kern, bias, out);
}